// AdaptiveCenterLoss_7267084665466
// MI455X (gfx1250) — compile-verified
//
#include <hip/hip_runtime.h>
#include <hip/hip_bf16.h>

typedef __attribute__((ext_vector_type(2))) float v2f;
typedef __attribute__((ext_vector_type(8))) float v8f;

#define D_DIM 256
#define K_STEPS (D_DIM / 4)          // 64 WMMA steps of K=4
#define WAVES_PER_BLOCK 8
#define ROWS_PER_BLOCK (WAVES_PER_BLOCK * 16)

// Per-wave: 16-row tile. acc += G_chunk x G_chunk^T with G = F - C(gathered).
// diag(acc) after the K loop = per-row squared distance ||f - c||^2.
__global__ __launch_bounds__(256) void center_loss_partial_kernel(
    const float* __restrict__ features,
    const float* __restrict__ centers,
    const int*   __restrict__ labels,
    float*       __restrict__ partial)
{
    const int lane = threadIdx.x & 31;
    const int wave = threadIdx.x >> 5;
    const int rowBase = (blockIdx.x * WAVES_PER_BLOCK + wave) << 4;
    const int m    = lane & 15;       // matrix row this lane serves
    const int half = lane >> 4;       // 0: K=0..1, 1: K=2..3
    const int r    = rowBase + m;

    const int lbl = labels[r];
    const float* fp = features + (size_t)r   * D_DIM + (half << 1);
    const float* cp = centers  + (size_t)lbl * D_DIM + (half << 1);

    v8f acc = {0.f, 0.f, 0.f, 0.f, 0.f, 0.f, 0.f, 0.f};

#pragma unroll
    for (int t = 0; t < K_STEPS; ++t) {
        v2f f = *(const v2f*)(fp + t * 4);   // 8B-aligned global_load_b64
        v2f c = *(const v2f*)(cp + t * 4);   // centers hot in L2 (1 MB)
        v2f g = f - c;
        // A == B (register-identical layouts for A and A^T) -> Gram matrix.
        acc = __builtin_amdgcn_wmma_f32_16x16x4_f32(
            /*neg_a=*/false, g, /*neg_b=*/false, g,
            /*c_mod=*/(short)0, acc, /*reuse_a=*/false, /*reuse_b=*/false);
    }

    // Extract diagonal: row j (0..7) at VGPR j lane j; row j+8 at VGPR j lane j+24.
    float v = 0.f;
#pragma unroll
    for (int j = 0; j < 8; ++j) {
        v += ((lane == j) || (lane == j + 24)) ? acc[j] : 0.0f;
    }

    // Wave32 reduction.
#pragma unroll
    for (int off = 16; off > 0; off >>= 1)
        v += __shfl_xor(v, off, 32);

    __shared__ float wsum[WAVES_PER_BLOCK];
    if (lane == 0) wsum[wave] = v;
    __syncthreads();
    if (threadIdx.x == 0) {
        float s = 0.f;
#pragma unroll
        for (int w = 0; w < WAVES_PER_BLOCK; ++w) s += wsum[w];
        partial[blockIdx.x] = s;
    }
}

// Deterministic final reduction of per-block partials; writes mean.
__global__ __launch_bounds__(256) void center_loss_final_kernel(
    const float* __restrict__ partial, float* __restrict__ out,
    int n, float invB)
{
    float v = 0.f;
    for (int i = threadIdx.x; i < n; i += 256) v += partial[i];
#pragma unroll
    for (int off = 16; off > 0; off >>= 1)
        v += __shfl_xor(v, off, 32);

    __shared__ float wsum[8];
    const int lane = threadIdx.x & 31;
    const int wave = threadIdx.x >> 5;
    if (lane == 0) wsum[wave] = v;
    __syncthreads();
    if (threadIdx.x == 0) {
        float t = 0.f;
#pragma unroll
        for (int w = 0; w < 8; ++w) t += wsum[w];
        out[0] = t * invB;
    }
}

extern "C" void kernel_launch(void* const* d_in, const int* in_sizes, int n_in,
                              void* d_out, int out_size, void* d_ws, size_t ws_size,
                              hipStream_t stream) {
    const float* features = (const float*)d_in[0];
    const float* centers  = (const float*)d_in[1];
    const int*   labels   = (const int*)d_in[2];

    const int B = in_sizes[2];              // 131072
    const int nBlocks = B / ROWS_PER_BLOCK; // 1024

    float* partial = (float*)d_ws;          // nBlocks floats of scratch

    center_loss_partial_kernel<<<nBlocks, 256, 0, stream>>>(
        features, centers, labels, partial);

    center_loss_final_kernel<<<1, 256, 0, stream>>>(
        partial, (float*)d_out, nBlocks, 1.0f / (float)B);
}